// SAGELANet_21071109554391
// MI455X (gfx1250) — compile-verified
//
#include <hip/hip_runtime.h>
#include <math.h>

typedef __attribute__((ext_vector_type(2))) float v2f;
typedef __attribute__((ext_vector_type(8))) float v8f;

#define Bn   4
#define Nn   10000
#define En   320000
#define Cn   64
#define OUTn 64
#define Kn   128   // 2*C

// ---------------------------------------------------------------------------
// Kernel 0: zero the workspace (agg[B*N*C] followed by deg[N]), float4 wide.
// ---------------------------------------------------------------------------
__global__ void zero_ws_kernel(float* __restrict__ p, int n4) {
    int i = blockIdx.x * blockDim.x + threadIdx.x;
    float4 z = {0.f, 0.f, 0.f, 0.f};
    for (; i < n4; i += gridDim.x * blockDim.x)
        reinterpret_cast<float4*>(p)[i] = z;
}

// ---------------------------------------------------------------------------
// Kernel 1: edge scatter. One wave32 per (edge, batch); lane owns channels
// 2*lane, 2*lane+1. Gate dot product reduced with 5 xor-shuffles. All gathers
// and atomic adds are L2-resident (X and agg are ~10 MB each, L2 = 192 MB).
// ---------------------------------------------------------------------------
__global__ void edge_kernel(const float* __restrict__ X,
                            const long long* __restrict__ ei,
                            const float* __restrict__ ew,
                            const float* __restrict__ ampw,
                            const float* __restrict__ gw,
                            const float* __restrict__ gb,
                            float* __restrict__ agg,
                            float* __restrict__ deg) {
    long long gtid = (long long)blockIdx.x * blockDim.x + threadIdx.x;
    long long wave = gtid >> 5;
    int lane = threadIdx.x & 31;
    if (wave >= (long long)En * Bn) return;   // uniform per wave

    int e = (int)(wave >> 2);   // /B  (B==4)
    int b = (int)(wave & 3);
    int src = (int)ei[e];
    int dst = (int)ei[En + e];
    float w = ew[e];

    const float2 xj  = reinterpret_cast<const float2*>(X + ((size_t)b * Nn + src) * Cn)[lane];
    const float2 xi  = reinterpret_cast<const float2*>(X + ((size_t)b * Nn + dst) * Cn)[lane];
    const float2 gwi = reinterpret_cast<const float2*>(gw)[lane];        // gate_w[0:64]
    const float2 gwj = reinterpret_cast<const float2*>(gw + Cn)[lane];   // gate_w[64:128]

    float part = xi.x * gwi.x + xi.y * gwi.y + xj.x * gwj.x + xj.y * gwj.y;
#pragma unroll
    for (int off = 16; off > 0; off >>= 1)
        part += __shfl_xor(part, off, 32);

    float z    = part + w * gw[2 * Cn] + gb[0];
    float lamb = 1.f / (1.f + __expf(-z));
    float s    = w * lamb;

    const float2 aw = reinterpret_cast<const float2*>(ampw)[lane];
    float* dp = agg + ((size_t)b * Nn + dst) * Cn + 2 * lane;
    atomicAdd(dp,     s * aw.x * xj.x);
    atomicAdd(dp + 1, s * aw.y * xj.y);
    if (b == 0 && lane == 0)
        atomicAdd(deg + dst, 1.0f);
}

// ---------------------------------------------------------------------------
// Kernel 2: agg /= max(deg, 1), in place, float4 wide (all 16 lanes of a
// float4 group share one node since C==64 is a multiple of 4).
// ---------------------------------------------------------------------------
__global__ void norm_kernel(float* __restrict__ agg, const float* __restrict__ deg) {
    int n4 = Bn * Nn * Cn / 4;
    int i = blockIdx.x * blockDim.x + threadIdx.x;
    for (; i < n4; i += gridDim.x * blockDim.x) {
        int n = (i >> 4) % Nn;            // (i*4 / C) % N, C==64
        float d = deg[n];
        float inv = 1.f / (d > 1.f ? d : 1.f);
        float4 v = reinterpret_cast<float4*>(agg)[i];
        v.x *= inv; v.y *= inv; v.z *= inv; v.w *= inv;
        reinterpret_cast<float4*>(agg)[i] = v;
    }
}

// ---------------------------------------------------------------------------
// Kernel 3: out = [X | agg] @ W + bias via V_WMMA_F32_16X16X4_F32.
// 256 threads = 8 waves; each wave owns one 16-row M tile and all four
// 16-col N tiles (4 x v8f accumulators). Weight staged in LDS in a
// PAIR-INTERLEAVED layout:  Wl[(k>>1)*128 + col*2 + (k&1)]
// so each lane's B fragment {W[k][col], W[k+1][col]} is one contiguous,
// 8B-aligned ds_load_b64 straight into an even register pair (no repack
// moves). A fragment (16x4 f32 per ISA layout): lanes 0-15 hold K=k0..k0+1
// (contiguous float2), lanes 16-31 hold K=k0+2..k0+3. C/D: VGPR r -> M=r
// (lanes<16) / M=r+8 (lanes>=16), N = lane&15. K loop split into an X half
// and an agg half so base pointers are loop-invariant (no cndmask selects).
// ---------------------------------------------------------------------------
__global__ void gemm_wmma_kernel(const float* __restrict__ X,
                                 const float* __restrict__ agg,
                                 const float* __restrict__ W,
                                 const float* __restrict__ bias,
                                 float* __restrict__ out) {
    __shared__ float Wl[Kn * OUTn];   // 32 KB, pair-interleaved
    int tid = threadIdx.x;
    for (int i = tid; i < Kn * OUTn; i += 256) {
        int k = i >> 6;        // 0..127
        int c = i & 63;        // 0..63
        Wl[(k >> 1) * 128 + c * 2 + (k & 1)] = W[i];
    }
    __syncthreads();

    int wave  = tid >> 5;
    int lane  = tid & 31;
    int mtile = blockIdx.x * 8 + wave;
    if (mtile >= (Bn * Nn) / 16) return;   // wave-uniform skip; EXEC stays all-1

    int lmod  = lane & 15;
    int lhalf = lane >> 4;                 // 0 / 1
    int khalf = lhalf * 2;                 // 0 for lanes 0-15, 2 for lanes 16-31
    int arow  = mtile * 16 + lmod;
    const float* xrow  = X   + (size_t)arow * Cn;
    const float* grow  = agg + (size_t)arow * Cn;
    // per-lane B base: K-pair selected by lane half, column by lane&15
    const float* Bbase = Wl + lhalf * 128 + lmod * 2;

    v8f acc[4];
#pragma unroll
    for (int t = 0; t < 4; ++t) {
        float bb = bias[t * 16 + lmod];
#pragma unroll
        for (int r = 0; r < 8; ++r) acc[t][r] = bb;
    }

    // K = 0..63 : X half of h
#pragma unroll 4
    for (int k0 = 0; k0 < Cn; k0 += 4) {
        v2f a = *reinterpret_cast<const v2f*>(xrow + k0 + khalf);
        const float* bp = Bbase + (k0 >> 1) * 128;
#pragma unroll
        for (int t = 0; t < 4; ++t) {
            v2f bm = *reinterpret_cast<const v2f*>(bp + t * 32);
            acc[t] = __builtin_amdgcn_wmma_f32_16x16x4_f32(
                false, a, false, bm, (short)0, acc[t], false, false);
        }
    }

    // K = 64..127 : agg half of h
#pragma unroll 4
    for (int k0 = 0; k0 < Cn; k0 += 4) {
        v2f a = *reinterpret_cast<const v2f*>(grow + k0 + khalf);
        const float* bp = Bbase + ((k0 + Cn) >> 1) * 128;
#pragma unroll
        for (int t = 0; t < 4; ++t) {
            v2f bm = *reinterpret_cast<const v2f*>(bp + t * 32);
            acc[t] = __builtin_amdgcn_wmma_f32_16x16x4_f32(
                false, a, false, bm, (short)0, acc[t], false, false);
        }
    }

    int rbase = mtile * 16 + lhalf * 8;
#pragma unroll
    for (int t = 0; t < 4; ++t)
#pragma unroll
        for (int r = 0; r < 8; ++r)
            out[(size_t)(rbase + r) * OUTn + t * 16 + lmod] = acc[t][r];
}

// ---------------------------------------------------------------------------
extern "C" void kernel_launch(void* const* d_in, const int* in_sizes, int n_in,
                              void* d_out, int out_size, void* d_ws, size_t ws_size,
                              hipStream_t stream) {
    const float*     X    = (const float*)d_in[0];
    const long long* ei   = (const long long*)d_in[1];
    const float*     ew   = (const float*)d_in[2];
    const float*     ampw = (const float*)d_in[3];
    const float*     gw   = (const float*)d_in[4];
    const float*     gb   = (const float*)d_in[5];
    const float*     W    = (const float*)d_in[6];
    const float*     bias = (const float*)d_in[7];
    float*           out  = (float*)d_out;

    float* agg = (float*)d_ws;                       // B*N*C floats (10.24 MB)
    float* deg = agg + (size_t)Bn * Nn * Cn;         // N floats

    // 1) zero agg + deg
    int zn4 = (Bn * Nn * Cn + Nn) / 4;
    zero_ws_kernel<<<1024, 256, 0, stream>>>((float*)d_ws, zn4);

    // 2) edge scatter: one wave per (edge, batch)
    long long total_waves = (long long)En * Bn;          // 1.28M waves
    long long total_thr   = total_waves * 32;
    int eblocks = (int)((total_thr + 255) / 256);        // 160000
    edge_kernel<<<eblocks, 256, 0, stream>>>(X, ei, ew, ampw, gw, gb, agg, deg);

    // 3) degree normalize
    norm_kernel<<<1024, 256, 0, stream>>>(agg, deg);

    // 4) WMMA GEMM: [X | agg] @ W + bias
    int mtiles  = (Bn * Nn) / 16;        // 2500
    int gblocks = (mtiles + 7) / 8;      // 313
    gemm_wmma_kernel<<<gblocks, 256, 0, stream>>>(X, agg, W, bias, out);
}